// NeuromorphicMemory_50964081934729
// MI455X (gfx1250) — compile-verified
//
#include <hip/hip_runtime.h>
#include <hip/hip_bf16.h>
#include <math.h>

// Problem constants (match reference)
#define B_DIM 4096
#define H_DIM 2048
#define CAP   65536
#define AGE_LIM 100

// Masked-sum decomposition
#define C_CHUNK 1024
#define N_CHUNK (CAP / C_CHUNK)   // 64 chunks
#define H_TILE  128               // columns per block (8 waves x 16)
#define H_BLOCKS (H_DIM / H_TILE) // 16

typedef __attribute__((ext_vector_type(2))) float v2f;
typedef __attribute__((ext_vector_type(8))) float v8f;

// Workspace layout (float offsets). Total ~820 KB.
#define OFF_SCAL 0                      // [1]=cond, [2]=count, [3]=inv
#define OFF_NORM 16                     // 4096 per-row norms
#define OFF_ROW  (OFF_NORM + B_DIM)     // 2048 column means
#define OFF_MASK (OFF_ROW + H_DIM)      // 65536 mask floats
#define OFF_YP   (OFF_MASK + CAP)       // 64*2048 partials
#define OFF_Y    (OFF_YP + N_CHUNK * H_DIM) // 2048 final vector

// ---------------- K1: per-batch-row L2 norms ----------------
__global__ __launch_bounds__(256) void k_rownorm(const float* __restrict__ x,
                                                 float* __restrict__ norms) {
    __shared__ float red[256];
    const float* row = x + (size_t)blockIdx.x * H_DIM;
    float s = 0.f;
    #pragma unroll 4
    for (int i = threadIdx.x; i < H_DIM; i += 256) { float v = row[i]; s += v * v; }
    red[threadIdx.x] = s; __syncthreads();
    for (int off = 128; off > 0; off >>= 1) {
        if (threadIdx.x < off) red[threadIdx.x] += red[threadIdx.x + off];
        __syncthreads();
    }
    if (threadIdx.x == 0) norms[blockIdx.x] = sqrtf(red[0]);
}

// ---------------- K2: column mean of x over batch ----------------
__global__ __launch_bounds__(256) void k_colmean(const float* __restrict__ x,
                                                 float* __restrict__ rowmean) {
    __shared__ float red[256];
    const int col = blockIdx.x * 64 + (threadIdx.x & 63);
    const int r   = threadIdx.x >> 6;   // 0..3
    float s = 0.f;
    for (int b = r; b < B_DIM; b += 4) s += x[(size_t)b * H_DIM + col];
    red[threadIdx.x] = s; __syncthreads();
    if (threadIdx.x < 64) {
        rowmean[col] = (red[threadIdx.x] + red[threadIdx.x + 64] +
                        red[threadIdx.x + 128] + red[threadIdx.x + 192]) * (1.0f / B_DIM);
    }
}

// ---------------- K3: cond, mask array, count (single block, deterministic) --
__global__ __launch_bounds__(256) void k_mask(const int* __restrict__ ages,
                                              const int* __restrict__ ptr_p,
                                              const float* __restrict__ strength_p,
                                              const float* __restrict__ norms,
                                              float* __restrict__ scal,
                                              float* __restrict__ mask) {
    __shared__ float red[256];
    // deterministic reduction of 4096 row norms
    float s = 0.f;
    for (int i = threadIdx.x; i < B_DIM; i += 256) s += norms[i];
    red[threadIdx.x] = s; __syncthreads();
    for (int off = 128; off > 0; off >>= 1) {
        if (threadIdx.x < off) red[threadIdx.x] += red[threadIdx.x + off];
        __syncthreads();
    }
    const float strength = red[0] * (1.0f / B_DIM);
    const int cond = strength > strength_p[0];
    const int ptr  = ptr_p[0];
    __syncthreads();

    float cnt = 0.f;
    for (int c = threadIdx.x; c < CAP; c += 256) {
        int age = ages[c];
        if (cond && c == ptr) age = 0;
        age += 1;
        float m = (age < AGE_LIM) ? 1.0f : 0.0f;
        mask[c] = m;
        cnt += m;
    }
    red[threadIdx.x] = cnt; __syncthreads();
    for (int off = 128; off > 0; off >>= 1) {
        if (threadIdx.x < off) red[threadIdx.x] += red[threadIdx.x + off];
        __syncthreads();
    }
    if (threadIdx.x == 0) {
        float count = red[0];
        scal[1] = cond ? 1.0f : 0.0f;
        scal[2] = count;
        scal[3] = 1.0f / fmaxf(count, 1.0f);
    }
}

// ---------------- K4: masked column-sum of bank via V_WMMA_F32_16X16X4_F32 --
// D[m][n] = sum_k A[m][k]*B[k][n] + C.  A[m][k] = mask[c+k] (broadcast over m),
// B[k][n] = bank[c+k][h0+n].  Per-lane layout (ISA 7.12.2, 32-bit 16x4 / 4x16):
// lanes 0-15 carry K0/K1 in VGPR0/1, lanes 16-31 carry K2/K3.
// D VGPR0, lanes 0-15 = row M=0, N=lane -> the desired partial sums.
__global__ __launch_bounds__(256) void k_masked_sum(const float* __restrict__ bank,
                                                    const float* __restrict__ mask,
                                                    float* __restrict__ yp) {
    const int wave = threadIdx.x >> 5;
    const int lane = threadIdx.x & 31;
    const int hi   = lane >> 4;                    // 0: K0/K1, 1: K2/K3
    const int col  = blockIdx.x * H_TILE + wave * 16 + (lane & 15);
    const int c0   = blockIdx.y * C_CHUNK;
    const int c1   = c0 + C_CHUNK;

    v8f acc = {};
    for (int c = c0; c < c1; c += 4) {
        const int cr = c + 2 * hi;
        // mask pair (8-byte aligned: cr is even)
        v2f a;
        const float2 mp = *reinterpret_cast<const float2*>(mask + cr);
        a[0] = mp.x; a[1] = mp.y;
        // bank tile elements for this lane
        v2f b;
        b[0] = bank[(size_t)cr * H_DIM + col];
        b[1] = bank[(size_t)(cr + 1) * H_DIM + col];
        acc = __builtin_amdgcn_wmma_f32_16x16x4_f32(
            /*neg_a=*/false, a, /*neg_b=*/false, b,
            /*c_mod=*/(short)0, acc, /*reuse_a=*/false, /*reuse_b=*/false);
    }
    if (lane < 16) yp[(size_t)blockIdx.y * H_DIM + col] = acc[0];
}

// ---------------- K5a: combine partials + pointer-row correction -------------
__global__ __launch_bounds__(256) void k_final(const float* __restrict__ bank,
                                               const float* __restrict__ rowmean,
                                               const float* __restrict__ scal,
                                               const int* __restrict__ ptr_p,
                                               const float* __restrict__ yp,
                                               float* __restrict__ y) {
    const int h = blockIdx.x * 256 + threadIdx.x;
    const float cond = scal[1], count = scal[2], inv = scal[3];
    const int ptr = ptr_p[0];
    float s = 0.f;
    #pragma unroll 8
    for (int ch = 0; ch < N_CHUNK; ++ch) s += yp[(size_t)ch * H_DIM + h];
    s += cond * (rowmean[h] - bank[(size_t)ptr * H_DIM + h]);
    y[h] = (count > 0.f) ? s * inv : 0.f;
}

// ---------------- K5b: broadcast 2048-vector to [B,1,H] output --------------
__global__ __launch_bounds__(256) void k_broadcast(const float4* __restrict__ yf4,
                                                   float4* __restrict__ out4,
                                                   int n4) {
    const int i = blockIdx.x * 256 + threadIdx.x;
    if (i < n4) out4[i] = yf4[i & (H_DIM / 4 - 1)];
}

extern "C" void kernel_launch(void* const* d_in, const int* in_sizes, int n_in,
                              void* d_out, int out_size, void* d_ws, size_t ws_size,
                              hipStream_t stream) {
    const float* x        = (const float*)d_in[0];
    const float* bank     = (const float*)d_in[1];
    const int*   ages     = (const int*)d_in[2];
    const int*   ptr_p    = (const int*)d_in[3];
    const float* strength = (const float*)d_in[4];
    // d_in[5] (forgetting_rate) does not affect the returned value.
    float* out = (float*)d_out;
    float* ws  = (float*)d_ws;   // needs ~820 KB

    float* scal = ws + OFF_SCAL;
    float* norm = ws + OFF_NORM;
    float* row  = ws + OFF_ROW;
    float* mask = ws + OFF_MASK;
    float* yp   = ws + OFF_YP;
    float* y    = ws + OFF_Y;

    // K1: 4096 row norms
    k_rownorm<<<B_DIM, 256, 0, stream>>>(x, norm);
    // K2: column mean (2048 cols, 64 per block)
    k_colmean<<<H_DIM / 64, 256, 0, stream>>>(x, row);
    // K3: cond / mask / count
    k_mask<<<1, 256, 0, stream>>>(ages, ptr_p, strength, norm, scal, mask);
    // K4: WMMA masked column-sum of the 512 MB bank
    dim3 grid4(H_BLOCKS, N_CHUNK);
    k_masked_sum<<<grid4, 256, 0, stream>>>(bank, mask, yp);
    // K5a: reduce partials + correction + scale
    k_final<<<H_DIM / 256, 256, 0, stream>>>(bank, row, scal, ptr_p, yp, y);
    // K5b: broadcast to output (float4)
    const int n4 = (B_DIM * H_DIM) / 4;
    k_broadcast<<<(n4 + 255) / 256, 256, 0, stream>>>(
        (const float4*)y, (float4*)out, n4);
}